// LI_no_Spike_66443144069282
// MI455X (gfx1250) — compile-verified
//
#include <hip/hip_runtime.h>
#include <cstdint>

// LI (leaky-integrate, no spike) scan over T for x:[B,F,T] f32.
// Memory-bound (0.25 flop/byte -> ~22us at 23.3 TB/s): optimize purely for
// HBM bandwidth with the CDNA5 async global<->LDS DMA engine (ASYNCcnt) and
// an LDS transpose so the per-row sequential scan still gets fully coalesced
// 128-bit global transactions. Streaming data is marked non-temporal.

namespace {

constexpr int kB      = 128;
constexpr int kF      = 1024;
constexpr int kT      = 500;
constexpr int kRows   = 256;              // rows per workgroup == blockDim.x
constexpr int kTC     = 64;               // time-chunk length (floats)
constexpr int kStride = kTC + 4;          // LDS row stride (floats): 272B, 16B aligned
constexpr int kChunks = (kT + kTC - 1) / kTC;            // 8
constexpr int kLast   = kT - (kChunks - 1) * kTC;        // 52 (multiple of 4)

__device__ __forceinline__ void async_load_b128(uint32_t lds, unsigned long long gaddr) {
  // VDST = LDS byte address, VADDR = 64-bit global address; NT: read-once stream
  asm volatile("global_load_async_to_lds_b128 %0, %1, off th:TH_LOAD_NT"
               :: "v"(lds), "v"(gaddr) : "memory");
}

__device__ __forceinline__ void async_store_b128(unsigned long long gaddr, uint32_t lds) {
  asm volatile("global_store_async_from_lds_b128 %0, %1, off th:TH_STORE_NT"
               :: "v"(gaddr), "v"(lds) : "memory");
}

__device__ __forceinline__ void wait_async_0() {
  asm volatile("s_wait_asynccnt 0x0" ::: "memory");
}

// Stage a [kRows x LEN] slab into LDS. Lane mapping: tv = tid&15 (16B vec slot,
// loop-invariant), row = (tid>>4) + 16*i. Lanes 0..15 cover one row's 256B run,
// lanes 16..31 the next -> fully coalesced. The 16-step burst fully unrolls
// with constant address increments. Tail (LEN=52) predicates tv<13 per lane.
template <int LEN>
__device__ __forceinline__ void issue_chunk_loads(const float* xbase, uint32_t ldsBase,
                                                  int t0, int tid) {
  const int tv = tid & 15;
  const int r0 = tid >> 4;
  if (tv * 4 < LEN) {
    unsigned long long g = (unsigned long long)(uintptr_t)
        (xbase + (long long)r0 * kT + t0 + tv * 4);
    uint32_t l = ldsBase + (uint32_t)(r0 * kStride + tv * 4) * 4u;
#pragma unroll
    for (int i = 0; i < 16; ++i) {
      async_load_b128(l, g);
      g += 16ull * kT * sizeof(float);        // +16 rows
      l += 16u * kStride * (uint32_t)sizeof(float);
    }
  }
}

template <int LEN>
__device__ __forceinline__ void issue_chunk_stores(float* obase, uint32_t ldsBase,
                                                   int t0, int tid) {
  const int tv = tid & 15;
  const int r0 = tid >> 4;
  if (tv * 4 < LEN) {
    unsigned long long g = (unsigned long long)(uintptr_t)
        (obase + (long long)r0 * kT + t0 + tv * 4);
    uint32_t l = ldsBase + (uint32_t)(r0 * kStride + tv * 4) * 4u;
#pragma unroll
    for (int i = 0; i < 16; ++i) {
      async_store_b128(g, l);
      g += 16ull * kT * sizeof(float);
      l += 16u * kStride * (uint32_t)sizeof(float);
    }
  }
}

// Sequential leaky integration on one LDS row, in place. Fully unrolled:
// LEN/4 x (ds_load_b128 -> 4 chained v_fmac_f32 -> ds_store_b128).
template <int LEN>
__device__ __forceinline__ float scan_row(float* row, float tc, float u) {
#pragma unroll
  for (int t = 0; t < LEN; t += 4) {
    float4 v = *reinterpret_cast<float4*>(row + t);
    v.x = fmaf(tc, u,   v.x);
    v.y = fmaf(tc, v.x, v.y);
    v.z = fmaf(tc, v.y, v.z);
    v.w = fmaf(tc, v.z, v.w);
    u = v.w;
    *reinterpret_cast<float4*>(row + t) = v;
  }
  return u;
}

__global__ void li_scan_kernel(const float* __restrict__ x,
                               const float* __restrict__ tau,
                               float* __restrict__ out) {
  extern __shared__ float smem[];           // 2 buffers of kRows*kStride floats
  const int tid = threadIdx.x;
  const long long rowBase = (long long)blockIdx.x * kRows;

  // Per-thread decay constant (constrain 'forward': clamp into [0,1])
  const int f = (int)((rowBase + tid) & (kF - 1));
  float tc = tau[f];
  tc = fminf(fmaxf(tc, 0.0f), 1.0f);

  float* buf0 = smem;
  float* buf1 = smem + kRows * kStride;
  const uint32_t lds0 = (uint32_t)(uintptr_t)buf0;  // low 32 bits = LDS byte addr
  const uint32_t lds1 = (uint32_t)(uintptr_t)buf1;

  const float* xrow = x + rowBase * kT;
  float*       orow = out + rowBase * kT;

  // Prime the pipeline: chunk 0 -> buffer 0
  issue_chunk_loads<kTC>(xrow, lds0, 0, tid);

  float carry = 0.0f;
  int p = 0;

  // 7 full 64-float chunks
  for (int c = 0; c < kChunks - 1; ++c) {
    // load(c) into buf p complete AND store(c-1) out of buf p^1 drained
    // (async LDS-side accesses may reorder -> must wait before refilling).
    wait_async_0();
    __syncthreads();

    // Prefetch chunk c+1 into the other buffer; overlaps with the scan.
    if (c + 1 < kChunks - 1)
      issue_chunk_loads<kTC>(xrow, p ? lds0 : lds1, (c + 1) * kTC, tid);
    else
      issue_chunk_loads<kLast>(xrow, p ? lds0 : lds1, (kChunks - 1) * kTC, tid);

    float* row = (p ? buf1 : buf0) + tid * kStride;
    carry = scan_row<kTC>(row, tc, carry);

    __syncthreads();  // publish scan results to the lanes that DMA them out
    issue_chunk_stores<kTC>(orow, p ? lds1 : lds0, c * kTC, tid);
    p ^= 1;
  }

  // Epilogue: 52-float tail chunk
  wait_async_0();
  __syncthreads();
  {
    float* row = (p ? buf1 : buf0) + tid * kStride;
    carry = scan_row<kLast>(row, tc, carry);
  }
  __syncthreads();
  issue_chunk_stores<kLast>(orow, p ? lds1 : lds0, (kChunks - 1) * kTC, tid);
  wait_async_0();
}

} // namespace

extern "C" void kernel_launch(void* const* d_in, const int* in_sizes, int n_in,
                              void* d_out, int out_size, void* d_ws, size_t ws_size,
                              hipStream_t stream) {
  (void)in_sizes; (void)n_in; (void)d_ws; (void)ws_size; (void)out_size;
  const float* x   = (const float*)d_in[0];
  const float* tau = (const float*)d_in[1];
  float*       out = (float*)d_out;

  constexpr int rowsTotal = kB * kF;                       // 131072 rows
  dim3 grid(rowsTotal / kRows);                            // 512 workgroups
  dim3 block(kRows);                                       // 256 threads = 8 waves
  const size_t shmem = 2ull * kRows * kStride * sizeof(float);  // 139264 B -> 2 WG/WGP
  li_scan_kernel<<<grid, block, shmem, stream>>>(x, tau, out);
}